// CombinedLoraA_59459527246479
// MI455X (gfx1250) — compile-verified
//
#include <hip/hip_runtime.h>

typedef __attribute__((ext_vector_type(2))) float v2f;
typedef __attribute__((ext_vector_type(4))) float v4f;
typedef __attribute__((ext_vector_type(8))) float v8f;

#define KDIM 4096
#define RDIM 64
#define ADAPT_STRIDE (KDIM * RDIM)

// One block per combo c; 4 waves (wave32); wave w computes out[c, 16w..16w+15]
// as the diagonal of a 16x16 WMMA accumulation over K=4096 in f32.
__global__ __launch_bounds__(128, 1) void lora_diag_wmma_kernel(
    const float* __restrict__ x,      // (512, 1, 4096) f32
    const int*   __restrict__ xids,   // (320*64,) int32
    const int*   __restrict__ wids,   // (320,) int32
    const float* __restrict__ A,      // (80, 4096, 64) f32
    float*       __restrict__ out)    // (320, 1, 64) f32
{
    const int c    = blockIdx.x;
    const int wave = threadIdx.x >> 5;        // r-tile index 0..3
    const int lane = threadIdx.x & 31;
    const int n    = lane & 15;               // row M for A-frag, col N for B-frag
    const int half = lane >> 4;               // 0: WMMA k=0,1  1: WMMA k=2,3

    const int w    = wids[c];
    const int xrow = xids[c * RDIM + wave * 16 + n];

    // Lane's x pointer: row xrow, starting at k-offset 4*half.
    const float* xptr = x + (size_t)xrow * KDIM + 4 * half;
    // Lane's A pointer: adapter w, k-offset 4*half, column (wave*16 + n).
    const float* aptr = A + (size_t)w * ADAPT_STRIDE
                          + (size_t)(4 * half) * RDIM
                          + (wave * 16 + n);

    v8f acc0 = {};
    v8f acc1 = {};

    // Each chunk covers 8 problem-K values: lane-half h consumes k0+4h..k0+4h+3.
    // K-order inside the WMMA is permuted (legal: dot product is order-invariant)
    // so the x fragment is one contiguous b128 load per lane per chunk.
    #pragma unroll 4
    for (int k0 = 0; k0 < KDIM; k0 += 8) {
        v4f xa = *(const v4f*)(xptr + k0);
        const float* ap = aptr + (size_t)k0 * RDIM;
        float b0 = ap[0 * RDIM];
        float b1 = ap[1 * RDIM];
        float b2 = ap[2 * RDIM];
        float b3 = ap[3 * RDIM];

        v2f a0 = { xa.x, xa.y };
        v2f a1 = { xa.z, xa.w };
        v2f bb0 = { b0, b1 };
        v2f bb1 = { b2, b3 };

        // 8-arg form: (neg_a, A, neg_b, B, c_mod, C, reuse_a, reuse_b)
        acc0 = __builtin_amdgcn_wmma_f32_16x16x4_f32(
                   false, a0, false, bb0, (short)0, acc0, false, false);
        acc1 = __builtin_amdgcn_wmma_f32_16x16x4_f32(
                   false, a1, false, bb1, (short)0, acc1, false, false);
    }

    // Diagonal extraction from the 16x16 f32 C/D layout:
    //   lanes 0-15:  VGPR v holds (M=v,   N=lane)   -> diag m=lane for lane<8
    //   lanes 16-31: VGPR v holds (M=v+8, N=lane-16)-> diag m=lane-16 for lane>=24
    int m = -1;
    if (lane < 8)        m = lane;
    else if (lane >= 24) m = lane - 16;

    if (m >= 0) {
        const int idx = (m >= 8) ? (m - 8) : m;   // VGPR slot holding the diagonal
        float v = acc0[0] + acc1[0];
        #pragma unroll
        for (int j = 1; j < 8; ++j) {
            float cj = acc0[j] + acc1[j];
            v = (idx == j) ? cj : v;
        }
        out[(size_t)c * RDIM + wave * 16 + m] = v;
    }
}

extern "C" void kernel_launch(void* const* d_in, const int* in_sizes, int n_in,
                              void* d_out, int out_size, void* d_ws, size_t ws_size,
                              hipStream_t stream) {
    const float* x    = (const float*)d_in[0];
    const int*   xids = (const int*)d_in[1];
    const int*   wids = (const int*)d_in[2];
    const float* A    = (const float*)d_in[3];
    float*       out  = (float*)d_out;

    const int C = in_sizes[2];   // COMBINED_BS = 320
    dim3 grid(C), block(128);
    hipLaunchKernelGGL(lora_diag_wmma_kernel, grid, block, 0, stream,
                       x, xids, wids, A, out);
}